// SFNO_66451734003805
// MI455X (gfx1250) — compile-verified
//
#include <hip/hip_runtime.h>

// ---------------------------------------------------------------------------
// SFNO spectral block for MI455X (gfx1250, wave32).
// fp32 end-to-end: total traffic ~0.9GB @ 23.3TB/s => ~40us floor; the
// per-mode channel GEMM is only ~0.9 GFLOP (AI ~2.8 flop/B) -> memory bound,
// so fp32 WMMA 16x16x4 + DMA data movement (TDM / async-to-LDS) is the
// right configuration, not low-precision matrix paths.
// Workspace requirement: 178,257,920 bytes.
// ---------------------------------------------------------------------------

constexpr int Bn  = 4;    // batch
constexpr int CIc = 20;   // in channels
constexpr int COc = 20;   // out channels
constexpr int NX  = 128;
constexpr int NY  = 128;
constexpr int NT  = 32;
constexpr int MX  = 32;   // kept x modes per corner
constexpr int MY  = 32;
constexpr int MT  = 17;   // = NT/2+1 (all temporal rfft modes kept)
constexpr int NXH = 64;   // total kept x modes (two corners)
constexpr int NYH = 64;

#define TWO_PI_F 6.28318530717958647692f

typedef float v2f __attribute__((ext_vector_type(2)));
typedef float v8f __attribute__((ext_vector_type(8)));
typedef unsigned int u32x4 __attribute__((ext_vector_type(4)));
typedef int i32x4 __attribute__((ext_vector_type(4)));
typedef int i32x8 __attribute__((ext_vector_type(8)));

// async-to-LDS builtins want int-vector pointers in AS1 (global) / AS3 (LDS)
typedef int avec2 __attribute__((vector_size(8)));
typedef int avec4 __attribute__((vector_size(16)));
#define GP2(p) ((__attribute__((address_space(1))) avec2*)(p))
#define LP2(p) ((__attribute__((address_space(3))) avec2*)(p))
#define GP4(p) ((__attribute__((address_space(1))) avec4*)(p))
#define LP4(p) ((__attribute__((address_space(3))) avec4*)(p))

// -------- CDNA5 data-mover availability ------------------------------------
#if __has_builtin(__builtin_amdgcn_global_load_async_to_lds_b128) && \
    __has_builtin(__builtin_amdgcn_global_load_async_to_lds_b64)  && \
    __has_builtin(__builtin_amdgcn_s_wait_asynccnt)
#define HAS_ASYNC 1
#else
#define HAS_ASYNC 0
#endif

#if __has_builtin(__builtin_amdgcn_tensor_load_to_lds) && \
    __has_builtin(__builtin_amdgcn_s_wait_tensorcnt)
#define HAS_TDM 1
#else
#define HAS_TDM 0
#endif

#if HAS_TDM
// Issue one 2D TDM tile load: `rows` rows of `row_words` DWORDs, row pitch
// `stride_words` DWORDs, packed contiguously into LDS at byte addr lds_addr.
// D# layout per CDNA5 ISA (data_size=4B,
// count=1, type=2, no pad/iterate/gather, groups 2&3 zero => 2D).
static __device__ __forceinline__
void tdm_load_2d(unsigned lds_addr, const void* gaddr,
                 unsigned rows, unsigned row_words,
                 unsigned long long stride_words)
{
    unsigned long long ga = (unsigned long long)gaddr;
    u32x4 g0;
    g0[0] = 1u;                                            // count=1
    g0[1] = lds_addr;                                      // lds_addr [63:32]
    g0[2] = (unsigned)(ga & 0xFFFFFFFFull);                // global_addr lo
    g0[3] = (unsigned)((ga >> 32) & 0x01FFFFFFull)         // global_addr hi
          | 0x80000000u;                                   // type=2
    i32x8 g1;
    g1[0] = (int)(2u << 16);                               // data_size=4B
    g1[1] = (int)((row_words & 0xFFFFu) << 16);            // tensor_dim0 lo16
    g1[2] = (int)((row_words >> 16) | ((rows & 0xFFFFu) << 16));
    g1[3] = (int)((rows >> 16) | ((row_words & 0xFFFFu) << 16)); // tile_dim0
    g1[4] = (int)(rows & 0xFFFFu);                         // tile_dim1 (dim2=0)
    g1[5] = (int)(stride_words & 0xFFFFFFFFull);           // dim0_stride lo
    g1[6] = (int)((stride_words >> 32) & 0xFFFFull);       // dim0_stride hi
    g1[7] = 0;
    i32x4 gz = {0, 0, 0, 0};
#if __clang_major__ >= 23
    i32x8 gz8 = {0, 0, 0, 0, 0, 0, 0, 0};
    __builtin_amdgcn_tensor_load_to_lds(g0, g1, gz, gz, gz8, 0);
#else
    __builtin_amdgcn_tensor_load_to_lds(g0, g1, gz, gz, 0);
#endif
}
static __device__ __forceinline__ unsigned lds_byte_addr(const void* p)
{
    // generic pointers into LDS carry the LDS byte offset in the low 32 bits
    return (unsigned)(unsigned long long)p;
}
#endif

// ---------------------------------------------------------------------------
// K1: fused t-rfft (32 -> 17 modes) + y-DFT (128 -> 64 kept modes).
// One block per (b,ci,x) row. Input slab is 128*32 contiguous floats.
// Out layout: (B*CI, NX, NYH, MT) complex.
// ---------------------------------------------------------------------------
__global__ __launch_bounds__(128)
void k_fwd_ty(const float* __restrict__ v, float2* __restrict__ outB)
{
    __shared__ float  s_in[NY * NT];       // 16384 B
    __shared__ float2 s_t [NY * MT];       // 17408 B
    __shared__ float2 tw32 [NT];
    __shared__ float2 tw128[NX];

    const int tid = threadIdx.x;
    const int blk = blockIdx.x;
    const int x   = blk & (NX - 1);
    const int bi  = blk >> 7;              // b*CI + ci

    const float* src = v + ((size_t)bi * NX + x) * (size_t)(NY * NT);
#if HAS_ASYNC
    for (int idx = tid; idx < (NY * NT) / 4; idx += 128)
        __builtin_amdgcn_global_load_async_to_lds_b128(
            GP4(src + idx * 4), LP4(s_in + idx * 4), 0, 0);
#endif
    for (int j = tid; j < NT; j += 128) {
        float sn, cs; __sincosf(-TWO_PI_F * (float)j / (float)NT, &sn, &cs);
        tw32[j] = make_float2(cs, sn);
    }
    for (int j = tid; j < NX; j += 128) {
        float sn, cs; __sincosf(-TWO_PI_F * (float)j / (float)NX, &sn, &cs);
        tw128[j] = make_float2(cs, sn);
    }
#if HAS_ASYNC
    __builtin_amdgcn_s_wait_asynccnt(0);
#else
    for (int idx = tid; idx < NY * NT; idx += 128) s_in[idx] = src[idx];
#endif
    __syncthreads();

    // t-rfft (real -> complex), direct 17x32 DFT
    for (int idx = tid; idx < NY * MT; idx += 128) {
        const int y = idx / MT, k = idx - y * MT;
        const float* r = &s_in[y * NT];
        float sr = 0.f, si = 0.f;
        #pragma unroll 8
        for (int n = 0; n < NT; ++n) {
            const float2 tw = tw32[(k * n) & (NT - 1)];
            sr += r[n] * tw.x; si += r[n] * tw.y;
        }
        s_t[idx] = make_float2(sr, si);
    }
    __syncthreads();

    // y-DFT, keep modes [0,32) U [96,128)
    float2* dst = outB + ((size_t)bi * NX + x) * (size_t)(NYH * MT);
    for (int idx = tid; idx < NYH * MT; idx += 128) {
        const int m = idx / MT, k = idx - m * MT;
        const int f = (m < MY) ? m : (m + NY - 2 * MY);
        float sr = 0.f, si = 0.f;
        #pragma unroll 8
        for (int n = 0; n < NY; ++n) {
            const float2 a  = s_t[n * MT + k];
            const float2 tw = tw128[(f * n) & (NX - 1)];
            sr += a.x * tw.x - a.y * tw.y;
            si += a.x * tw.y + a.y * tw.x;
        }
        dst[idx] = make_float2(sr, si);
    }
}

// ---------------------------------------------------------------------------
// K2: x-DFT (128 -> 64 kept modes). One block per (b*ci, ym).
// In: (B*CI, NX, NYH, MT)  Out: (B*CI, NXH, NYH, MT)
// ---------------------------------------------------------------------------
__global__ __launch_bounds__(128)
void k_fwd_x(const float2* __restrict__ inB, float2* __restrict__ outC)
{
    __shared__ float2 s_in[NX * MT];       // 17408 B
    __shared__ float2 tw128[NX];

    const int tid = threadIdx.x;
    const int blk = blockIdx.x;
    const int ym  = blk & (NYH - 1);
    const int bi  = blk >> 6;

#if HAS_ASYNC
    for (int idx = tid; idx < NX * MT; idx += 128) {
        const int n = idx / MT, k = idx - n * MT;
        __builtin_amdgcn_global_load_async_to_lds_b64(
            GP2(inB + (((size_t)bi * NX + n) * NYH + ym) * MT + k),
            LP2(s_in + idx), 0, 0);
    }
#endif
    for (int j = tid; j < NX; j += 128) {
        float sn, cs; __sincosf(-TWO_PI_F * (float)j / (float)NX, &sn, &cs);
        tw128[j] = make_float2(cs, sn);
    }
#if HAS_ASYNC
    __builtin_amdgcn_s_wait_asynccnt(0);
#else
    for (int idx = tid; idx < NX * MT; idx += 128) {
        const int n = idx / MT, k = idx - n * MT;
        s_in[idx] = inB[(((size_t)bi * NX + n) * NYH + ym) * MT + k];
    }
#endif
    __syncthreads();

    for (int idx = tid; idx < NXH * MT; idx += 128) {
        const int m = idx / MT, k = idx - m * MT;
        const int f = (m < MX) ? m : (m + NX - 2 * MX);
        float sr = 0.f, si = 0.f;
        #pragma unroll 8
        for (int n = 0; n < NX; ++n) {
            const float2 a  = s_in[n * MT + k];
            const float2 tw = tw128[(f * n) & (NX - 1)];
            sr += a.x * tw.x - a.y * tw.y;
            si += a.x * tw.y + a.y * tw.x;
        }
        outC[(((size_t)bi * NXH + m) * NYH + ym) * MT + k] = make_float2(sr, si);
    }
}

// ---------------------------------------------------------------------------
// K3: spectral channel mix via V_WMMA_F32_16X16X4_F32.
// Complex GEMM per mode: out[b,o] = sum_i vh[b,i]*w[i,o], packed as real GEMM
//   A' (4 x 40):  A'[b,2i]=ar, A'[b,2i+1]=ai   (== float2 layout directly)
//   W' (40 x 40): col 2o: [br; -bi], col 2o+1: [bi; br]
//   C  (4 x 40):  col 2o = cr, col 2o+1 = ci
// N padded to 48 -> 3 N-tiles, K=40 -> 10 K-steps: 30 WMMA per mode.
// One block per (xm, ym, quadrant); wave w handles t = w, w+4, ...
// Weight tile (400 rows x 136B) and vh tile (80 rows x 136B) are staged by
// the Tensor Data Mover (one descriptor each, issued by waves 0 and 1).
// ---------------------------------------------------------------------------
static __device__ __forceinline__
v2f make_b(const float2* __restrict__ s_w, int i_idx, int col, int t)
{
    v2f r;
    const int o  = col >> 1;
    const int qp = col & 1;
    const float2 ww = s_w[(i_idx * COc + o) * MT + t];
    r.x = qp ? ww.y : ww.x;    // K even row (p=0): re-col->br, im-col->bi
    r.y = qp ? ww.x : -ww.y;   // K odd  row (p=1): re-col->-bi, im-col->br
    return r;
}

__global__ __launch_bounds__(128)
void k_spectral_wmma(const float2* __restrict__ vh,
                     const float2* __restrict__ w0,
                     const float2* __restrict__ w1,
                     const float2* __restrict__ w2,
                     const float2* __restrict__ w3,
                     float2* __restrict__ oh)
{
    __shared__ float2 s_v[Bn * CIc * MT];    // 10,880 B
    __shared__ float2 s_w[CIc * COc * MT];   // 54,400 B (total 65,280 < 64KiB)

    const int tid = threadIdx.x;
    const int xm = blockIdx.x, ym = blockIdx.y, q = blockIdx.z;
    const float2* w = (q == 0) ? w0 : (q == 1) ? w1 : (q == 2) ? w2 : w3;
    const int ix = q & 1, iy = q >> 1;
    const int xt = ix * MX + xm;             // kept-mode index in [0,64)
    const int yt = iy * MY + ym;

    const size_t wmode = ((size_t)xm * MY + ym) * MT;
    const int wave = tid >> 5;
    const int lane = tid & 31;

#if HAS_TDM
    if (wave == 0) {
        // weights: 400 rows of 17 float2 (34 DWORDs), pitch 17408 float2
        tdm_load_2d(lds_byte_addr(s_w), (const void*)(w + wmode),
                    (unsigned)(CIc * COc), 34u,
                    (unsigned long long)(MX * MY * MT) * 2ull);
        __builtin_amdgcn_s_wait_tensorcnt(0);
    } else if (wave == 1) {
        // vh: 80 rows of 17 float2, pitch NXH*NYH*MT float2
        tdm_load_2d(lds_byte_addr(s_v),
                    (const void*)(vh + ((size_t)xt * NYH + yt) * MT),
                    (unsigned)(Bn * CIc), 34u,
                    (unsigned long long)(NXH * NYH * MT) * 2ull);
        __builtin_amdgcn_s_wait_tensorcnt(0);
    }
#elif HAS_ASYNC
    for (int idx = tid; idx < CIc * COc * MT; idx += 128) {
        const int io = idx / MT, t = idx - io * MT;
        __builtin_amdgcn_global_load_async_to_lds_b64(
            GP2(w + (size_t)io * (MX * MY * MT) + wmode + t),
            LP2(s_w + idx), 0, 0);
    }
    for (int idx = tid; idx < Bn * CIc * MT; idx += 128) {
        const int bi = idx / MT, t = idx - bi * MT;
        __builtin_amdgcn_global_load_async_to_lds_b64(
            GP2(vh + (((size_t)bi * NXH + xt) * NYH + yt) * MT + t),
            LP2(s_v + idx), 0, 0);
    }
    __builtin_amdgcn_s_wait_asynccnt(0);
#else
    for (int idx = tid; idx < CIc * COc * MT; idx += 128) {
        const int io = idx / MT, t = idx - io * MT;
        s_w[idx] = w[(size_t)io * (MX * MY * MT) + wmode + t];
    }
    for (int idx = tid; idx < Bn * CIc * MT; idx += 128) {
        const int bi = idx / MT, t = idx - bi * MT;
        s_v[idx] = vh[(((size_t)bi * NXH + xt) * NYH + yt) * MT + t];
    }
#endif
    __syncthreads();

    const int row = lane & 15;      // A row (M) / B col (N)
    const int h   = lane >> 4;      // lane half: K = 4s + 2h + {0,1}

    for (int t = wave; t < MT; t += 4) {
        v8f c0 = {}, c1 = {}, c2 = {};
        #pragma unroll
        for (int s = 0; s < 10; ++s) {
            const int i_idx = 2 * s + h;          // channel index of lane's K pair
            // A tile (16x4): rows >= Bn are zero padding
            v2f a; a.x = 0.f; a.y = 0.f;
            if (row < Bn) {
                const float2 vv = s_v[(row * CIc + i_idx) * MT + t];
                a.x = vv.x;                       // K even -> ar
                a.y = vv.y;                       // K odd  -> ai
            }
            // B tiles (4x16 each), cols: [0,16), [16,32), [32,48) of W' (N=40)
            const v2f b0 = make_b(s_w, i_idx, row,      t);
            const v2f b1 = make_b(s_w, i_idx, row + 16, t);
            v2f b2; b2.x = 0.f; b2.y = 0.f;
            if (row < 8) b2 = make_b(s_w, i_idx, row + 32, t);

            c0 = __builtin_amdgcn_wmma_f32_16x16x4_f32(false, a, false, b0,
                                                       (short)0, c0, false, false);
            c1 = __builtin_amdgcn_wmma_f32_16x16x4_f32(false, a, false, b1,
                                                       (short)0, c1, false, false);
            c2 = __builtin_amdgcn_wmma_f32_16x16x4_f32(false, a, false, b2,
                                                       (short)0, c2, false, false);
        }
        // Write back: lanes 0..15 hold M rows 0..7 in vgprs 0..7 (rows 0..3 valid)
        if (lane < 16) {
            float* of = (float*)oh;
            #pragma unroll
            for (int bb = 0; bb < Bn; ++bb) {
                {   const int gc = row;          // tile 0
                    const int o = gc >> 1, p = gc & 1;
                    const size_t fo = ((((size_t)bb * COc + o) * NXH + xt) * NYH + yt) * MT + t;
                    of[fo * 2 + p] = c0[bb];
                }
                {   const int gc = row + 16;     // tile 1
                    const int o = gc >> 1, p = gc & 1;
                    const size_t fo = ((((size_t)bb * COc + o) * NXH + xt) * NYH + yt) * MT + t;
                    of[fo * 2 + p] = c1[bb];
                }
                if (row < 8) {                   // tile 2 (cols 32..39)
                    const int gc = row + 32;
                    const int o = gc >> 1, p = gc & 1;
                    const size_t fo = ((((size_t)bb * COc + o) * NXH + xt) * NYH + yt) * MT + t;
                    of[fo * 2 + p] = c2[bb];
                }
            }
        }
    }
}

// ---------------------------------------------------------------------------
// K4: x inverse (64 kept modes -> 128). One block per (b*co, ym).
// In: (B*CO, NXH, NYH, MT)  Out: (B*CO, NX, NYH, MT)
// ---------------------------------------------------------------------------
__global__ __launch_bounds__(128)
void k_inv_x(const float2* __restrict__ inD, float2* __restrict__ outE)
{
    __shared__ float2 s_in[NXH * MT];      // 8704 B
    __shared__ float2 tw128[NX];

    const int tid = threadIdx.x;
    const int blk = blockIdx.x;
    const int ym  = blk & (NYH - 1);
    const int bo  = blk >> 6;

#if HAS_ASYNC
    for (int idx = tid; idx < NXH * MT; idx += 128) {
        const int m = idx / MT, k = idx - m * MT;
        __builtin_amdgcn_global_load_async_to_lds_b64(
            GP2(inD + (((size_t)bo * NXH + m) * NYH + ym) * MT + k),
            LP2(s_in + idx), 0, 0);
    }
#endif
    for (int j = tid; j < NX; j += 128) {
        float sn, cs; __sincosf(-TWO_PI_F * (float)j / (float)NX, &sn, &cs);
        tw128[j] = make_float2(cs, sn);
    }
#if HAS_ASYNC
    __builtin_amdgcn_s_wait_asynccnt(0);
#else
    for (int idx = tid; idx < NXH * MT; idx += 128) {
        const int m = idx / MT, k = idx - m * MT;
        s_in[idx] = inD[(((size_t)bo * NXH + m) * NYH + ym) * MT + k];
    }
#endif
    __syncthreads();

    for (int idx = tid; idx < NX * MT; idx += 128) {
        const int X = idx / MT, k = idx - X * MT;
        float sr = 0.f, si = 0.f;
        #pragma unroll 8
        for (int m = 0; m < NXH; ++m) {
            const int f = (m < MX) ? m : (m + NX - 2 * MX);
            const float2 a  = s_in[m * MT + k];
            const float2 tw = tw128[(f * X) & (NX - 1)];  // e^{-i}; apply conj
            sr += a.x * tw.x + a.y * tw.y;
            si += a.y * tw.x - a.x * tw.y;
        }
        outE[(((size_t)bo * NX + X) * NYH + ym) * MT + k] = make_float2(sr, si);
    }
}

// ---------------------------------------------------------------------------
// K5: fused y inverse (64 -> 128) + t-irfft (17 -> 32 real, Hermitian C2R,
// uses Re of bins 0 and 16 only, like pocketfft). Applies 1/(NX*NY*NT).
// One block per (b*co, X). Out: (B, CO, NX, NY, NT) real.
// ---------------------------------------------------------------------------
__global__ __launch_bounds__(128)
void k_inv_yt(const float2* __restrict__ inE, float* __restrict__ out)
{
    __shared__ float2 s_in[NYH * MT];      // 8704 B
    __shared__ float2 s_y [NY * MT];       // 17408 B
    __shared__ float2 tw128[NX];
    __shared__ float2 tw32 [NT];

    const int tid = threadIdx.x;
    const int blk = blockIdx.x;
    const int X   = blk & (NX - 1);
    const int bo  = blk >> 7;

    const float2* src = inE + ((size_t)bo * NX + X) * (size_t)(NYH * MT);
#if HAS_ASYNC
    for (int idx = tid; idx < (NYH * MT) / 2; idx += 128)   // 544 x b128
        __builtin_amdgcn_global_load_async_to_lds_b128(
            GP4(src + idx * 2), LP4(s_in + idx * 2), 0, 0);
#endif
    for (int j = tid; j < NX; j += 128) {
        float sn, cs; __sincosf(-TWO_PI_F * (float)j / (float)NX, &sn, &cs);
        tw128[j] = make_float2(cs, sn);
    }
    for (int j = tid; j < NT; j += 128) {
        float sn, cs; __sincosf(-TWO_PI_F * (float)j / (float)NT, &sn, &cs);
        tw32[j] = make_float2(cs, sn);
    }
#if HAS_ASYNC
    __builtin_amdgcn_s_wait_asynccnt(0);
#else
    for (int idx = tid; idx < NYH * MT; idx += 128) s_in[idx] = src[idx];
#endif
    __syncthreads();

    // y inverse over 64 kept modes
    for (int idx = tid; idx < NY * MT; idx += 128) {
        const int y = idx / MT, k = idx - y * MT;
        float sr = 0.f, si = 0.f;
        #pragma unroll 8
        for (int m = 0; m < NYH; ++m) {
            const int f = (m < MY) ? m : (m + NY - 2 * MY);
            const float2 a  = s_in[m * MT + k];
            const float2 tw = tw128[(f * y) & (NX - 1)];
            sr += a.x * tw.x + a.y * tw.y;
            si += a.y * tw.x - a.x * tw.y;
        }
        s_y[idx] = make_float2(sr, si);
    }
    __syncthreads();

    // t-irfft (real output)
    const float scale = 1.0f / (float)(NX * NY * NT);
    float* dst = out + ((size_t)bo * NX + X) * (size_t)(NY * NT);
    for (int idx = tid; idx < NY * NT; idx += 128) {
        const int y = idx >> 5, n = idx & (NT - 1);
        const float2 X0  = s_y[y * MT + 0];
        const float2 X16 = s_y[y * MT + 16];
        float acc = X0.x + ((n & 1) ? -X16.x : X16.x);
        #pragma unroll
        for (int k = 1; k < 16; ++k) {
            const float2 a  = s_y[y * MT + k];
            const float2 tw = tw32[(k * n) & (NT - 1)];   // (cos, -sin)
            acc += 2.0f * (a.x * tw.x + a.y * tw.y);      // Re(a * e^{+i th})
        }
        dst[idx] = acc * scale;
    }
}

// ---------------------------------------------------------------------------
extern "C" void kernel_launch(void* const* d_in, const int* in_sizes, int n_in,
                              void* d_out, int out_size, void* d_ws, size_t ws_size,
                              hipStream_t stream)
{
    const float*  v  = (const float*)d_in[0];
    const float2* w0 = (const float2*)d_in[1];
    const float2* w1 = (const float2*)d_in[2];
    const float2* w2 = (const float2*)d_in[3];
    const float2* w3 = (const float2*)d_in[4];
    float* out = (float*)d_out;

    // Workspace layout (bytes). Needs 178,257,920 B total.
    //   B  (y-trunc spectrum): 80*128*64*17*8 = 89,128,960   @ 0
    //   C  (x-trunc spectrum): 80*64*64*17*8  = 44,564,480   @ 89,128,960
    //   D  (mixed spectrum)  : 44,564,480                    @ 133,693,440
    //   E  (x-expanded)      : 89,128,960                    @ 0 (reuses B, dead)
    char* ws = (char*)d_ws;
    float2* Bb = (float2*)(ws);
    float2* Cb = (float2*)(ws + 89128960);
    float2* Db = (float2*)(ws + 133693440);
    float2* Eb = (float2*)(ws);

    k_fwd_ty       <<<dim3(Bn * CIc * NX),  dim3(128), 0, stream>>>(v, Bb);
    k_fwd_x        <<<dim3(Bn * CIc * NYH), dim3(128), 0, stream>>>(Bb, Cb);
    k_spectral_wmma<<<dim3(MX, MY, 4),      dim3(128), 0, stream>>>(Cb, w0, w1, w2, w3, Db);
    k_inv_x        <<<dim3(Bn * COc * NYH), dim3(128), 0, stream>>>(Db, Eb);
    k_inv_yt       <<<dim3(Bn * COc * NX),  dim3(128), 0, stream>>>(Eb, out);
}